// DetectionLoss_18313740550445
// MI455X (gfx1250) — compile-verified
//
#include <hip/hip_runtime.h>
#include <hip/hip_bf16.h>
#include <math.h>

// ---------------------------------------------------------------------------
// YOLO detection loss for MI455X (gfx1250, wave32).
//
// Roofline: dominant cost is the objectness pass over 403,200 cells with a
// 340-byte stride (~52 MB of unique cachelines) -> ~2.2us at 23.3 TB/s.
// Everything else (3072 boxes x 85 channels) is noise. So the design is:
//   * never materialize the dense obj target:  BCE(x,1)-BCE(x,0) == -x,
//     so obj_sum = sum_all softplus(x) - sum_{unique pos cells} x
//   * dedup positive cells with a bitmask + atomicOr (matches .at[].max())
//   * wave-level f32 reduction via V_WMMA_F32_16X16X4_F32 (exact in f32):
//     A = ones(16x4), B = (partials, 0) -> D[m,n] = p[n] + p[n+16],
//     then 4 shfl_xor steps finish the 16-lane fold.
// ---------------------------------------------------------------------------

#define NUM_CLASSES 80
#define BATCH 16
#define MAXB 64

typedef __attribute__((ext_vector_type(2))) float v2f;
typedef __attribute__((ext_vector_type(8))) float v8f;

// workspace float slots
#define ACC_CLS   0   // [3] per-layer sum of masked class BCE
#define ACC_BOX   3   // [3] per-layer sum of masked CIoU loss
#define ACC_NPOS  6   // [3] per-layer positive count
#define ACC_POSX  9   // [3] per-layer sum of obj logits over UNIQUE pos cells
#define ACC_SP    12  // [3] per-layer sum of softplus(obj logit) over all cells
#define ACC_FLOATS 16

// dedup bitmask: 307200/32 + 76800/32 + 19200/32 = 9600 + 2400 + 600 words
#define FLAG_W0 9600
#define FLAG_W1 2400
#define FLAG_W2 600
#define FLAG_WORDS_TOTAL (FLAG_W0 + FLAG_W1 + FLAG_W2)

__device__ __forceinline__ float bce_logits(float x, float t) {
    // stable BCE-with-logits, matches jnp formulation
    return fmaxf(x, 0.0f) - x * t + log1pf(expf(-fabsf(x)));
}

__global__ void yolo_zero_kernel(unsigned int* __restrict__ ws, int n) {
    int i = blockIdx.x * blockDim.x + threadIdx.x;
    if (i < n) ws[i] = 0u;
}

// One thread per (layer, batch, box): anchor matching, class BCE, CIoU,
// and dedup'd objectness-logit collection for positive cells.
__global__ void yolo_box_kernel(const float* __restrict__ p3,
                                const float* __restrict__ p4,
                                const float* __restrict__ p5,
                                const float* __restrict__ boxes,
                                const int*   __restrict__ labels,
                                const unsigned char* __restrict__ valid, // JAX bool = 1 byte
                                const float* __restrict__ anchors,      // [3,3,2]
                                float* __restrict__ acc,
                                unsigned int* __restrict__ flags) {
    const int NBM = BATCH * MAXB;
    int idx = blockIdx.x * blockDim.x + threadIdx.x;
    if (idx >= 3 * NBM) return;
    int l = idx / NBM;
    int r = idx - l * NBM;          // flat (b, m)
    int b = r / MAXB;

    if (!valid[r]) return;

    const int gdim_tab[3] = {80, 40, 20};
    const float* pred = (l == 0) ? p3 : (l == 1) ? p4 : p5;
    const int g = gdim_tab[l];
    const float gf = (float)g;

    float x1 = boxes[r * 4 + 0];
    float y1 = boxes[r * 4 + 1];
    float x2 = boxes[r * 4 + 2];
    float y2 = boxes[r * 4 + 3];
    float cx = (x1 + x2) * 0.5f * gf;
    float cy = (y1 + y2) * 0.5f * gf;
    float w  = (x2 - x1) * gf;
    float h  = (y2 - y1) * gf;

    // wh-only IoU vs this layer's 3 anchors; first-max argmax (jnp.argmax)
    float best = -1.0f;
    int ba = 0;
    #pragma unroll
    for (int a = 0; a < 3; ++a) {
        float aw = anchors[l * 6 + a * 2 + 0];
        float ah = anchors[l * 6 + a * 2 + 1];
        float inter = fminf(w, aw) * fminf(h, ah);
        float uni = w * h + aw * ah - inter;
        float iou = inter / (uni + 1e-6f);
        if (iou > best) { best = iou; ba = a; }
    }
    if (!(best > 0.5f)) return;      // pos = valid & (best_iou > IOU_THR)

    int gx = (int)cx;  gx = gx < 0 ? 0 : (gx > g - 1 ? g - 1 : gx);
    int gy = (int)cy;  gy = gy < 0 ? 0 : (gy > g - 1 ? g - 1 : gy);

    long cell = ((long)(b * 3 + ba) * g + gy) * g + gx;
    const float* p = pred + cell * 85;

    // ---- class BCE vs one-hot ----
    int lab = labels[r];
    float cls = 0.0f;
    #pragma unroll 4
    for (int c = 0; c < NUM_CLASSES; ++c) {
        float t = (c == lab) ? 1.0f : 0.0f;
        cls += bce_logits(p[5 + c], t);
    }

    // ---- CIoU-style box loss (matches reference formula exactly) ----
    float pw = p[2], ph = p[3];
    float px1 = p[0] - pw * 0.5f, py1 = p[1] - ph * 0.5f;
    float px2 = p[0] + pw * 0.5f, py2 = p[1] + ph * 0.5f;
    float tx1 = cx - w * 0.5f, ty1 = cy - h * 0.5f;
    float tx2 = cx + w * 0.5f, ty2 = cy + h * 0.5f;

    float ix1 = fmaxf(px1, tx1), iy1 = fmaxf(py1, ty1);
    float ix2 = fminf(px2, tx2), iy2 = fminf(py2, ty2);
    float inter = fmaxf(ix2 - ix1, 0.0f) * fmaxf(iy2 - iy1, 0.0f);
    float a1 = (px2 - px1) * (py2 - py1);
    float a2 = (tx2 - tx1) * (ty2 - ty1);
    float iou = inter / (a1 + a2 - inter + 1e-7f);
    float pcx = (px1 + px2) * 0.5f, pcy = (py1 + py2) * 0.5f;
    float tcx = (tx1 + tx2) * 0.5f, tcy = (ty1 + ty2) * 0.5f;
    float cd = (pcx - tcx) * (pcx - tcx) + (pcy - tcy) * (pcy - tcy);
    float ex1 = fminf(px1, tx1), ey1 = fminf(py1, ty1);
    float ex2 = fmaxf(px2, tx2), ey2 = fmaxf(py2, ty2);
    float dd = (ex2 - ex1) * (ex2 - ex1) + (ey2 - ey1) * (ey2 - ey1);
    float bbox = 1.0f - (iou - cd / (dd + 1e-7f));

    atomicAdd(&acc[ACC_CLS + l], cls);
    atomicAdd(&acc[ACC_BOX + l], bbox);
    atomicAdd(&acc[ACC_NPOS + l], 1.0f);

    // ---- dedup objectness target (semantics of .at[...].max(m)) ----
    const int flag_off_tab[3] = {0, FLAG_W0, FLAG_W0 + FLAG_W1};
    unsigned int word = (unsigned int)flag_off_tab[l] + (unsigned int)(cell >> 5);
    unsigned int bit = 1u << ((unsigned int)cell & 31u);
    unsigned int old = atomicOr(&flags[word], bit);
    if (!(old & bit)) atomicAdd(&acc[ACC_POSX + l], p[4]);
}

// Strided softplus reduction over pred[...,4] for one pyramid level.
// Wave reduction through the XDL pipe: one f32 WMMA folds 32->16 exactly,
// shfl_xor finishes. EXEC is all-1s here (uniform control flow, blockDim 256).
__global__ void yolo_obj_kernel(const float* __restrict__ pred, int ncells,
                                float* __restrict__ acc, int slot) {
    float partial = 0.0f;
    int stride = gridDim.x * blockDim.x;
    for (int i = blockIdx.x * blockDim.x + threadIdx.x; i < ncells; i += stride) {
        float x = pred[(long)i * 85 + 4];
        partial += fmaxf(x, 0.0f) + log1pf(expf(-fabsf(x)));  // softplus = BCE(x, 0)
    }

#if __has_builtin(__builtin_amdgcn_wmma_f32_16x16x4_f32)
    // A = ones(16x4) (both A VGPRs = 1.0) ; B VGPR0 = partials, B VGPR1 = 0.
    // D[m,n] = partial[n] + partial[n+16] for every m -> lane L holds s[L&15].
    v2f av; av[0] = 1.0f; av[1] = 1.0f;
    v2f bv; bv[0] = partial; bv[1] = 0.0f;
    v8f cv = {};
    cv = __builtin_amdgcn_wmma_f32_16x16x4_f32(false, av, false, bv,
                                               (short)0, cv, false, false);
    float s = cv[0];
    #pragma unroll
    for (int m = 1; m < 16; m <<= 1) s += __shfl_xor(s, m, 32);
#else
    float s = partial;
    #pragma unroll
    for (int m = 1; m < 32; m <<= 1) s += __shfl_xor(s, m, 32);
#endif

    if ((threadIdx.x & 31) == 0) atomicAdd(&acc[slot], s);
}

__global__ void yolo_final_kernel(const float* __restrict__ acc,
                                  float* __restrict__ out) {
    if (blockIdx.x != 0 || threadIdx.x != 0) return;
    const float ncells_tab[3] = {16.0f * 3 * 80 * 80, 16.0f * 3 * 40 * 40,
                                 16.0f * 3 * 20 * 20};
    float cls = 0.0f, obj = 0.0f, box = 0.0f;
    #pragma unroll
    for (int l = 0; l < 3; ++l) {
        float npos = acc[ACC_NPOS + l];
        if (npos > 0.0f) {                       // has = (npos > 0)
            float denom = fmaxf(npos, 1.0f);
            cls += acc[ACC_CLS + l] / (denom * (float)NUM_CLASSES);
            box += acc[ACC_BOX + l] / denom;
            obj += (acc[ACC_SP + l] - acc[ACC_POSX + l]) / ncells_tab[l];
        }
    }
    out[0] = 0.5f * cls + 1.0f * obj + 0.05f * box;
}

extern "C" void kernel_launch(void* const* d_in, const int* in_sizes, int n_in,
                              void* d_out, int out_size, void* d_ws, size_t ws_size,
                              hipStream_t stream) {
    const float* p3      = (const float*)d_in[0];   // [16,3,80,80,85]
    const float* p4      = (const float*)d_in[1];   // [16,3,40,40,85]
    const float* p5      = (const float*)d_in[2];   // [16,3,20,20,85]
    const float* boxes   = (const float*)d_in[3];   // [16,64,4]
    const int*   labels  = (const int*)d_in[4];     // [16,64]
    const unsigned char* valid = (const unsigned char*)d_in[5]; // bool [16,64]
    const float* anchors = (const float*)d_in[6];   // [3,3,2]

    float* acc = (float*)d_ws;
    unsigned int* flags = (unsigned int*)d_ws + ACC_FLOATS;

    // 1) zero accumulators + dedup bitmask (~50 KB)
    int zero_words = ACC_FLOATS + FLAG_WORDS_TOTAL;
    yolo_zero_kernel<<<(zero_words + 255) / 256, 256, 0, stream>>>(
        (unsigned int*)d_ws, zero_words);

    // 2) per-box matching + cls/box losses + pos-cell collection
    int nbox = 3 * BATCH * MAXB;
    yolo_box_kernel<<<(nbox + 255) / 256, 256, 0, stream>>>(
        p3, p4, p5, boxes, labels, valid, anchors, acc, flags);

    // 3) softplus sums over every cell's objectness logit (bandwidth-bound)
    const int nc0 = BATCH * 3 * 80 * 80;
    const int nc1 = BATCH * 3 * 40 * 40;
    const int nc2 = BATCH * 3 * 20 * 20;
    int blk0 = (nc0 + 255) / 256; if (blk0 > 2048) blk0 = 2048;
    int blk1 = (nc1 + 255) / 256; if (blk1 > 2048) blk1 = 2048;
    int blk2 = (nc2 + 255) / 256; if (blk2 > 2048) blk2 = 2048;
    yolo_obj_kernel<<<blk0, 256, 0, stream>>>(p3, nc0, acc, ACC_SP + 0);
    yolo_obj_kernel<<<blk1, 256, 0, stream>>>(p4, nc1, acc, ACC_SP + 1);
    yolo_obj_kernel<<<blk2, 256, 0, stream>>>(p5, nc2, acc, ACC_SP + 2);

    // 4) combine with gains
    yolo_final_kernel<<<1, 32, 0, stream>>>(acc, (float*)d_out);
}